// Model_42167988912481
// MI455X (gfx1250) — compile-verified
//
#include <hip/hip_runtime.h>
#include <math.h>

typedef float v2f __attribute__((ext_vector_type(2)));
typedef float v8f __attribute__((ext_vector_type(8)));

// Problem constants
#define KB 64     // batch
#define KC 256    // channels
#define KD 64     // q/k dim
#define KN 128    // points

#define FACTOR     3.8197186342054885f   // 180/(pi*15) = 12/pi
#define RMUL       0.7498942093324559f   // 10000^(-1/32)
#define BN_SCALE   0.9999950000374997f   // 1/sqrt(1+1e-5)
#define INV_SQRT2  0.7071067811865475f

#define WMMA_F32(a, b, c) \
  __builtin_amdgcn_wmma_f32_16x16x4_f32(false, (a), false, (b), (short)0, (c), false, false)

// ---------------------------------------------------------------------------
// Generic batched GEMM, f32 WMMA 16x16x4, 2x2 tile blocking per wave.
//   C[b] = alpha * A[b] @ B[b]  (+ C[b] if accum)
// A: [M x K] row-major, lda; batch stride sA (0 => shared weights)
// B: [K x N] row-major, ldb; batch stride sB
// C: [M x N] row-major (or transposed store if transOut), ldc; stride sC
// alphaMode: 0 -> alpha=1 ; 1 -> alpha=*alphaPtr ; 2 -> alpha=1-*alphaPtr
// One wave computes one 32x32 macro-tile (4 accumulators): each A fragment is
// reused by 2 WMMAs and each B fragment by 2 WMMAs -> 2x arithmetic intensity.
// Requires M,N multiples of 32; K multiple of 4.
// ---------------------------------------------------------------------------
__global__ __launch_bounds__(128) void gemm_wmma_f32(
    const float* __restrict__ A, int lda, long long sA,
    const float* __restrict__ Bm, int ldb, long long sB,
    float* __restrict__ Cm, int ldc, long long sC,
    int M, int N, int K,
    const float* alphaPtr, int alphaMode, int accum, int transOut)
{
  const int lane = threadIdx.x & 31;
  const int wave = threadIdx.x >> 5;
  const int mtN = N >> 5;                      // macro-tiles along N
  const int mtM = M >> 5;
  const int mt  = blockIdx.x * 4 + wave;
  if (mt >= mtM * mtN) return;
  const int tm = (mt / mtN) << 5;
  const int tn = (mt % mtN) << 5;
  const int b  = blockIdx.y;

  const float* Ab = A  + (long long)b * sA;
  const float* Bb = Bm + (long long)b * sB;
  float*       Cb = Cm + (long long)b * sC;

  const int h   = lane >> 4;   // half-wave: 0 or 1
  const int r15 = lane & 15;

  v8f acc00 = {}, acc01 = {}, acc10 = {}, acc11 = {};
  for (int k0 = 0; k0 < K; k0 += 4) {
    const int ka = k0 + (h << 1);              // k for frag elem 0; +1 for elem 1
    // A fragments: rows tm+r15 and tm+16+r15 (layout: M=lane&15, k=v+2h)
    const float* ar0 = Ab + (long long)(tm + r15) * lda + ka;
    const float* ar1 = ar0 + (long long)16 * lda;
    v2f a0, a1;
    a0.x = ar0[0]; a0.y = ar0[1];
    a1.x = ar1[0]; a1.y = ar1[1];
    // B fragments: cols tn+r15 and tn+16+r15 (layout: N=lane&15, k=v+2h)
    const float* br0 = Bb + (long long)ka * ldb + tn + r15;
    const float* br1 = br0 + ldb;
    v2f b0, b1;
    b0.x = br0[0];  b0.y = br1[0];
    b1.x = br0[16]; b1.y = br1[16];

    acc00 = WMMA_F32(a0, b0, acc00);
    acc01 = WMMA_F32(a0, b1, acc01);
    acc10 = WMMA_F32(a1, b0, acc10);
    acc11 = WMMA_F32(a1, b1, acc11);
  }

  float alpha = 1.0f;
  if (alphaMode == 1) alpha = *alphaPtr;
  else if (alphaMode == 2) alpha = 1.0f - *alphaPtr;

  #pragma unroll
  for (int mi = 0; mi < 2; ++mi) {
    #pragma unroll
    for (int ni = 0; ni < 2; ++ni) {
      const v8f acc = (mi == 0) ? (ni == 0 ? acc00 : acc01)
                                : (ni == 0 ? acc10 : acc11);
      #pragma unroll
      for (int v = 0; v < 8; ++v) {
        const int row = tm + (mi << 4) + v + (h << 3);   // C layout: row=v+8h
        const int col = tn + (ni << 4) + r15;            //           col=lane&15
        const long long idx = transOut ? ((long long)col * ldc + row)
                                       : ((long long)row * ldc + col);
        float val = alpha * acc[v];
        if (accum) val += Cb[idx];
        Cb[idx] = val;
      }
    }
  }
}

// ---------------------------------------------------------------------------
// K2[b][c][m] += emb2[b][m][c]  where emb2 = PE(a_cross)@p2w.T + p2b
// block: 64 threads (one per c), grid: (N, B)
// ---------------------------------------------------------------------------
__global__ void emb2_add_kernel(const float* __restrict__ across,
                                const float* __restrict__ p2w,
                                const float* __restrict__ p2b,
                                float* __restrict__ K2)
{
  __shared__ float pe[KD];
  const int m = blockIdx.x, b = blockIdx.y, c = threadIdx.x;
  if (c < 32) {
    const float t  = across[b * KN + m] * FACTOR;
    const float dt = expf(-0.28782313662425576f * (float)c); // ln(1e4)/32 * j
    float s, co;
    sincosf(t * dt, &s, &co);
    pe[2 * c]     = s;
    pe[2 * c + 1] = co;
  }
  __syncthreads();
  float v = p2b[c];
  #pragma unroll 8
  for (int d = 0; d < KD; ++d) v += pe[d] * p2w[c * KD + d];
  K2[b * (KD * KN) + c * KN + m] += v;
}

// qb[b][n] = Q[b][n][:] . p1b   (grid: B, block: N)
__global__ void qb_kernel(const float* __restrict__ Q,
                          const float* __restrict__ p1b,
                          float* __restrict__ qb)
{
  const int b = blockIdx.x, n = threadIdx.x;
  const float* q = Q + b * (KN * KD) + n * KD;
  float s = 0.f;
  #pragma unroll 8
  for (int c = 0; c < KD; ++c) s += q[c] * p1b[c];
  qb[b * KN + n] = s;
}

// ---------------------------------------------------------------------------
// Pre-softmax energies (WMMA for both q@k GEMMs + transcendental angle bias):
//   E[b][n][m] = bw*(Q@Klrf + pe_bias(n,m)) + (1-bw)*(Q@K2)
// grid: (16, B), block: 128 (4 waves); wave per 16x16 tile of the 128x128 map
// ---------------------------------------------------------------------------
__global__ __launch_bounds__(128) void energy_wmma(
    const float* __restrict__ Q,      // [B][N][D]
    const float* __restrict__ Klrf,   // [B][D][N]
    const float* __restrict__ K2,     // [B][D][N]
    const float* __restrict__ q1,     // [B][N][D]
    const float* __restrict__ qb,     // [B][N]
    const float* __restrict__ aself,  // [B][N][N]
    const float* __restrict__ bwPtr,  // &beta_w[layer]
    float* __restrict__ E)            // [B][N][N]
{
  const int lane = threadIdx.x & 31;
  const int wave = threadIdx.x >> 5;
  const int tile = blockIdx.x * 4 + wave;     // 64 tiles (8x8)
  if (tile >= 64) return;
  const int tm = (tile >> 3) << 4;            // n-tile
  const int tn = (tile & 7) << 4;             // m-tile
  const int b  = blockIdx.y;

  const float* Qb = Q    + b * (KN * KD);
  const float* K1 = Klrf + b * (KD * KN);
  const float* Kp = K2   + b * (KD * KN);

  const int h   = lane >> 4;
  const int r15 = lane & 15;

  v8f acc1 = {};
  v8f acc2 = {};
  for (int k0 = 0; k0 < KD; k0 += 4) {
    const int ka = k0 + (h << 1);
    v2f av, b1, b2;
    av.x = Qb[(tm + r15) * KD + ka];
    av.y = Qb[(tm + r15) * KD + ka + 1];
    b1.x = K1[ka * KN + tn + r15];
    b1.y = K1[(ka + 1) * KN + tn + r15];
    b2.x = Kp[ka * KN + tn + r15];
    b2.y = Kp[(ka + 1) * KN + tn + r15];
    acc1 = WMMA_F32(av, b1, acc1);
    acc2 = WMMA_F32(av, b2, acc2);
  }

  const float bw = *bwPtr;
  const int m = tn + r15;
  #pragma unroll
  for (int v = 0; v < 8; ++v) {
    const int n = tm + v + (h << 3);
    const float* q1r = q1 + (b * KN + n) * KD;
    const float ang = aself[b * (KN * KN) + n * KN + m] * FACTOR;
    float bias = qb[b * KN + n];
    float dt = 1.0f;
    #pragma unroll 4
    for (int j = 0; j < 32; ++j) {
      float s, c;
      sincosf(ang * dt, &s, &c);
      bias += s * q1r[2 * j] + c * q1r[2 * j + 1];
      dt *= RMUL;
    }
    const float e = bw * (acc1[v] + bias) + (1.0f - bw) * acc2[v];
    E[b * (KN * KN) + n * KN + m] = e;
  }
}

// Row softmax in place over last dim.  grid: B, block: N (thread = row)
__global__ void softmax_rows(float* __restrict__ E)
{
  const int b = blockIdx.x, n = threadIdx.x;
  float* row = E + b * (KN * KN) + n * KN;
  float mx = row[0];
  for (int m = 1; m < KN; ++m) mx = fmaxf(mx, row[m]);
  float s = 0.f;
  for (int m = 0; m < KN; ++m) { const float e = expf(row[m] - mx); row[m] = e; s += e; }
  const float inv = 1.0f / s;
  for (int m = 0; m < KN; ++m) row[m] *= inv;
}

// Column renorm: E[:,n,m] /= (1e-12 + sum_n E).  grid: B, block: N (thread = column)
__global__ void col_renorm(float* __restrict__ E)
{
  const int b = blockIdx.x, m = threadIdx.x;
  float* Eb = E + b * (KN * KN);
  float s = 0.f;
  for (int n = 0; n < KN; ++n) s += Eb[n * KN + m];
  const float inv = 1.0f / (1e-12f + s);
  for (int n = 0; n < KN; ++n) Eb[n * KN + m] *= inv;
}

// R = X - R   (R holds x_r on entry)
__global__ void sub_kernel(const float* __restrict__ X, long long sx,
                           float* __restrict__ R)
{
  const int idx = blockIdx.x * blockDim.x + threadIdx.x;
  if (idx >= KB * KC * KN) return;
  const int b = idx >> 15;            // C*N = 32768 = 2^15
  const int r = idx & 32767;
  R[idx] = X[(long long)b * sx + r] - R[idx];
}

// out = X + gelu_exact( (H + bl) * BN_SCALE * g + bb )
__global__ void final_kernel(const float* __restrict__ X, long long sx,
                             const float* __restrict__ H,
                             const float* __restrict__ bl,
                             const float* __restrict__ g,
                             const float* __restrict__ bb,
                             float* __restrict__ out)   // d_out + layer*C*N
{
  const int idx = blockIdx.x * blockDim.x + threadIdx.x;
  if (idx >= KB * KC * KN) return;
  const int b = idx >> 15;
  const int r = idx & 32767;
  const int c = r >> 7;               // N = 128 = 2^7
  const float x = X[(long long)b * sx + r];
  float hv = (H[idx] + bl[c]) * BN_SCALE;
  hv = hv * g[c] + bb[c];
  const float ge = 0.5f * hv * (1.0f + erff(hv * INV_SQRT2));
  out[(long long)b * (4LL * KC * KN) + r] = x + ge;
}

// ---------------------------------------------------------------------------
extern "C" void kernel_launch(void* const* d_in, const int* in_sizes, int n_in,
                              void* d_out, int out_size, void* d_ws, size_t ws_size,
                              hipStream_t stream)
{
  const float* lrf     = (const float*)d_in[0];
  const float* pca     = (const float*)d_in[1];
  const float* a_self  = (const float*)d_in[2];
  const float* a_cross = (const float*)d_in[3];
  const float* Wq      = (const float*)d_in[4];   // [L,D,C]
  const float* Wk      = (const float*)d_in[5];   // [L,D,C]
  const float* Wv      = (const float*)d_in[6];   // [L,C,C]
  const float* Wvp     = (const float*)d_in[7];   // [L,C,C]
  const float* Wl      = (const float*)d_in[8];   // [L,C,C]
  const float* bl      = (const float*)d_in[9];   // [L,C]
  const float* bn_g    = (const float*)d_in[10];  // [L,C]
  const float* bn_b    = (const float*)d_in[11];  // [L,C]
  const float* p1w     = (const float*)d_in[12];  // [L,D,D]
  const float* p1b     = (const float*)d_in[13];  // [L,D]
  const float* p2w     = (const float*)d_in[14];  // [L,D,D]
  const float* p2b     = (const float*)d_in[15];  // [L,D]
  const float* beta_x  = (const float*)d_in[16];  // [L]
  const float* beta_w  = (const float*)d_in[17];  // [L]
  // d_in[18] = picked (unused by the forward output)

  float* out = (float*)d_out;

  // Workspace layout (floats)
  const long long QSZ = (long long)KB * KN * KD;   // 524288
  const long long ESZ = (long long)KB * KN * KN;   // 1048576
  const long long USZ = (long long)KB * KC * KN;   // 2097152
  float* ws   = (float*)d_ws;
  float* Qb   = ws;                    // [B][N][D]
  float* Klrf = Qb   + QSZ;            // [B][D][N]
  float* K2   = Klrf + QSZ;            // [B][D][N]
  float* q1b  = K2   + QSZ;            // [B][N][D]
  float* qbb  = q1b  + QSZ;            // [B][N]
  float* Eb   = qbb  + (KB * KN);      // [B][N][N]
  float* Uu   = Eb   + ESZ;            // [B][C][N]   (later aliased as H)
  float* Up   = Uu   + USZ;            // [B][C][N]
  float* Rb   = Up   + USZ;            // [B][C][N]
  if (ws_size < (size_t)((Rb + USZ) - ws) * sizeof(float)) return;

  const int EW = KB * KC * KN;         // elementwise count
  const dim3 blk128(128);
  // 32x32 macro-tiles, 4 waves per block
  #define GEMM_GRID(M, N) dim3(((((M) >> 5) * ((N) >> 5)) + 3) / 4, KB)

  for (int i = 0; i < 4; ++i) {
    const float* X  = (i == 0) ? lrf : (out + (long long)(i - 1) * KC * KN);
    const long long sx = (i == 0) ? (long long)KC * KN : 4LL * KC * KN;
    const float* wq = Wq + (long long)i * KD * KC;
    const float* wk = Wk + (long long)i * KD * KC;

    // Q[b] = (Wq @ X)^T  -> [N][D]
    gemm_wmma_f32<<<GEMM_GRID(KD, KN), blk128, 0, stream>>>(
        wq, KC, 0, X, KN, sx, Qb, KD, (long long)KN * KD,
        KD, KN, KC, nullptr, 0, 0, 1);
    // Klrf[b] = Wk @ X -> [D][N]
    gemm_wmma_f32<<<GEMM_GRID(KD, KN), blk128, 0, stream>>>(
        wk, KC, 0, X, KN, sx, Klrf, KN, (long long)KD * KN,
        KD, KN, KC, nullptr, 0, 0, 0);
    // K2[b] = Wq @ pca -> [D][N]
    gemm_wmma_f32<<<GEMM_GRID(KD, KN), blk128, 0, stream>>>(
        wq, KC, 0, pca, KN, (long long)KC * KN, K2, KN, (long long)KD * KN,
        KD, KN, KC, nullptr, 0, 0, 0);
    // K2 += emb2^T
    emb2_add_kernel<<<dim3(KN, KB), dim3(KD), 0, stream>>>(
        a_cross, p2w + (long long)i * KD * KD, p2b + (long long)i * KD, K2);
    // q1[b] = Q @ p1w -> [N][D]
    gemm_wmma_f32<<<GEMM_GRID(KN, KD), blk128, 0, stream>>>(
        Qb, KD, (long long)KN * KD, p1w + (long long)i * KD * KD, KD, 0,
        q1b, KD, (long long)KN * KD, KN, KD, KD, nullptr, 0, 0, 0);
    // qb[b][n] = Q . p1b
    qb_kernel<<<dim3(KB), dim3(KN), 0, stream>>>(Qb, p1b + (long long)i * KD, qbb);
    // pre-softmax energies
    energy_wmma<<<dim3(16, KB), blk128, 0, stream>>>(
        Qb, Klrf, K2, q1b, qbb, a_self, beta_w + i, Eb);
    // softmax over m, then column renorm over n
    softmax_rows<<<dim3(KB), dim3(KN), 0, stream>>>(Eb);
    col_renorm<<<dim3(KB), dim3(KN), 0, stream>>>(Eb);
    // Uu = X @ E ; Up = pca @ E
    gemm_wmma_f32<<<GEMM_GRID(KC, KN), blk128, 0, stream>>>(
        X, KN, sx, Eb, KN, (long long)KN * KN, Uu, KN, (long long)KC * KN,
        KC, KN, KN, nullptr, 0, 0, 0);
    gemm_wmma_f32<<<GEMM_GRID(KC, KN), blk128, 0, stream>>>(
        pca, KN, (long long)KC * KN, Eb, KN, (long long)KN * KN,
        Up, KN, (long long)KC * KN, KC, KN, KN, nullptr, 0, 0, 0);
    // R = beta_x*Wv@Uu + (1-beta_x)*Wvp@Up   (= x_r)
    gemm_wmma_f32<<<GEMM_GRID(KC, KN), blk128, 0, stream>>>(
        Wv + (long long)i * KC * KC, KC, 0, Uu, KN, (long long)KC * KN,
        Rb, KN, (long long)KC * KN, KC, KN, KC, beta_x + i, 1, 0, 0);
    gemm_wmma_f32<<<GEMM_GRID(KC, KN), blk128, 0, stream>>>(
        Wvp + (long long)i * KC * KC, KC, 0, Up, KN, (long long)KC * KN,
        Rb, KN, (long long)KC * KN, KC, KN, KC, beta_x + i, 2, 1, 0);
    // R = X - R
    sub_kernel<<<dim3((EW + 255) / 256), dim3(256), 0, stream>>>(X, sx, Rb);
    // H = Wl @ R   (H aliases Uu, which is free now)
    gemm_wmma_f32<<<GEMM_GRID(KC, KN), blk128, 0, stream>>>(
        Wl + (long long)i * KC * KC, KC, 0, Rb, KN, (long long)KC * KN,
        Uu, KN, (long long)KC * KN, KC, KN, KC, nullptr, 0, 0, 0);
    // out_slice = X + gelu(bn(H + bl))
    final_kernel<<<dim3((EW + 255) / 256), dim3(256), 0, stream>>>(
        X, sx, Uu, bl + (long long)i * KC, bn_g + (long long)i * KC,
        bn_b + (long long)i * KC, out + (long long)i * KC * KN);
  }
  #undef GEMM_GRID
}